// FlowMambaRotCell_54820962566301
// MI455X (gfx1250) — compile-verified
//
#include <hip/hip_runtime.h>
#include <hip/hip_bf16.h>

typedef __attribute__((ext_vector_type(2))) float v2f;
typedef __attribute__((ext_vector_type(8))) float v8f;
typedef int v4i __attribute__((vector_size(16)));

#define B_ 4
#define V_ 8
#define D_ 64
#define N_ 16
#define H_ 32
#define W_ 32
#define HW 1024
#define PI_D 3.14159265358979323846

static __device__ __forceinline__ int imin(int a, int b) { return a < b ? a : b; }
static __device__ __forceinline__ int imax(int a, int b) { return a > b ? a : b; }

// ---------------------------------------------------------------------------
// Kernel 1: GroupNorm over u_t -> u_norm (workspace)
// 16 blocks = (b, group); group = 16 channels x 32x32
// ---------------------------------------------------------------------------
__global__ __launch_bounds__(256) void gn_kernel(const float* __restrict__ u,
                                                 const float* __restrict__ gn_w,
                                                 const float* __restrict__ gn_b,
                                                 float* __restrict__ u_norm) {
    __shared__ float s_sum[256];
    __shared__ float s_sq[256];
    const int bg = blockIdx.x;
    const int b = bg >> 2, g = bg & 3;
    const float* base = u + ((size_t)b * D_ + g * 16) * HW;
    float sum = 0.f, sq = 0.f;
    for (int i = threadIdx.x; i < 16 * HW; i += 256) {
        float x = base[i];
        sum += x;
        sq += x * x;
    }
    s_sum[threadIdx.x] = sum;
    s_sq[threadIdx.x] = sq;
    __syncthreads();
    for (int o = 128; o > 0; o >>= 1) {
        if (threadIdx.x < o) {
            s_sum[threadIdx.x] += s_sum[threadIdx.x + o];
            s_sq[threadIdx.x] += s_sq[threadIdx.x + o];
        }
        __syncthreads();
    }
    const float inv_n = 1.0f / (16.0f * HW);
    float mu = s_sum[0] * inv_n;
    float var = s_sq[0] * inv_n - mu * mu;
    float rstd = rsqrtf(var + 1e-5f);
    float* ob = u_norm + ((size_t)b * D_ + g * 16) * HW;
    for (int i = threadIdx.x; i < 16 * HW; i += 256) {
        int c = g * 16 + (i >> 10);
        ob[i] = (base[i] - mu) * rstd * gn_w[c] + gn_b[c];
    }
}

// ---------------------------------------------------------------------------
// Kernel 2: fused 3x3 convs (delta 64ch + B 16ch + C 16ch) as WMMA f32 GEMM.
// GEMM: Y[oc, pix] = sum_k Wm[oc, k] * Im[k, pix],  k = kk*64 + ci (kk=3*ky+kx)
// grid = 256 pixel-tiles (16 consecutive pixels each), block = 192 (6 waves),
// wave w owns oc-tile w (oc 0..95). A = weights, B = implicit im2col.
// ---------------------------------------------------------------------------
__global__ __launch_bounds__(192) void conv_wmma_kernel(
    const float* __restrict__ u_norm, const float* __restrict__ wd,
    const float* __restrict__ bd, const float* __restrict__ wB,
    const float* __restrict__ wC, const float* __restrict__ dt_bias_p,
    float* __restrict__ delta, float* __restrict__ Bv, float* __restrict__ Cv) {
    const int tid = threadIdx.x;
    const int wave = tid >> 5;
    const int lane = tid & 31;
    const int hi = lane >> 4;   // which half-wave (selects K pair)
    const int m = lane & 15;    // A-row (oc within tile) and B-col (pixel)

    const int gpix = blockIdx.x * 16 + m;
    const int bb = gpix >> 10;
    const int rem = gpix & 1023;
    const int py = rem >> 5;
    const int px = rem & 31;

    // select weight tensor + local out-channel for this wave's oc-tile
    const float* wsel;
    int oc_off;
    if (wave < 4) { wsel = wd; oc_off = wave * 16; }
    else if (wave == 4) { wsel = wB; oc_off = 0; }
    else { wsel = wC; oc_off = 0; }
    const float* wrow = wsel + (size_t)(oc_off + m) * 576;  // OIHW flat: oc*576 + ci*9 + kk
    const float* ub = u_norm + (size_t)bb * D_ * HW;

    v8f c = {0.f, 0.f, 0.f, 0.f, 0.f, 0.f, 0.f, 0.f};

    for (int ks = 0; ks < 144; ++ks) {
        const int ka = ks * 4 + hi * 2;   // even; ci even, ci+1 same kk
        const int kk = ka >> 6;
        const int ci = ka & 63;
        const int dy = kk / 3 - 1;
        const int dx = kk - (kk / 3) * 3 - 1;

        v2f a;
        a.x = wrow[ci * 9 + kk];
        a.y = wrow[(ci + 1) * 9 + kk];

        v2f bf;
        const int yy = py + dy, xx = px + dx;
        if (yy >= 0 && yy < H_ && xx >= 0 && xx < W_) {
            const float* ip = ub + ci * HW + yy * W_ + xx;
            bf.x = ip[0];
            bf.y = ip[HW];
        } else {
            bf.x = 0.f;
            bf.y = 0.f;
        }
        c = __builtin_amdgcn_wmma_f32_16x16x4_f32(false, a, false, bf,
                                                  (short)0, c, false, false);
    }

    const float dtb = dt_bias_p[0];
#pragma unroll
    for (int r = 0; r < 8; ++r) {
        const int oc = wave * 16 + r + 8 * hi;  // global out-channel (0..95)
        const float val = c[r];
        if (oc < 64) {
            float xv = val + bd[oc] + dtb;
            float sp = (xv > 20.f) ? xv : log1pf(expf(xv));
            sp = fminf(fmaxf(sp, 1e-4f), 5.f);
            delta[((size_t)bb * D_ + oc) * HW + rem] = sp;
        } else if (oc < 80) {
            Bv[((size_t)bb * N_ + (oc - 64)) * HW + rem] = val;
        } else {
            Cv[((size_t)bb * N_ + (oc - 80)) * HW + rem] = val;
        }
    }
}

// ---------------------------------------------------------------------------
// Kernel 3: transport (rotation bilinear) + SSM state update + y reduction.
// One block per (b,v,d). 16 n-planes of s_prev staged to LDS (64 KB) via the
// CDNA5 async global->LDS engine (ASYNCcnt) when available, bilinear gather
// from LDS, fused A_bar/B_bar, s_new store + y dot-reduce over n.
// ---------------------------------------------------------------------------
__global__ __launch_bounds__(256) void state_kernel(
    const float* __restrict__ u_t, const float* __restrict__ s_prev,
    const float* __restrict__ log_A_real, const float* __restrict__ D_param,
    const float* __restrict__ delta, const float* __restrict__ Bv,
    const float* __restrict__ Cv, float* __restrict__ out_y,
    float* __restrict__ out_s) {
    extern __shared__ float sP[];  // 16 * 1024 floats = 64 KB

    const int blk = blockIdx.x;
    const int d = blk & 63;
    const int v = (blk >> 6) & 7;
    const int b = blk >> 9;

    const size_t sbase = (size_t)((b * V_ + v) * D_ + d) * (N_ * HW);

#if __has_builtin(__builtin_amdgcn_global_load_async_to_lds_b128)
    // Direct global -> LDS async copy (no VGPR round-trip); ASYNCcnt-tracked.
    {
        typedef v4i __attribute__((address_space(1))) * gvec_p;
        typedef v4i __attribute__((address_space(3))) * lvec_p;
        const float* gsrc = s_prev + sbase;
        for (int i = threadIdx.x; i < (N_ * HW) / 4; i += 256) {
            gvec_p gp = (gvec_p)(uintptr_t)(gsrc + i * 4);
            // Generic LDS pointers carry the LDS byte offset in addr[31:0].
            lvec_p lp = (lvec_p)(uint32_t)(uintptr_t)(sP + i * 4);
            __builtin_amdgcn_global_load_async_to_lds_b128(gp, lp, 0, 0);
        }
        asm volatile("s_wait_asynccnt 0x0" ::: "memory");
    }
#else
    {
        const float4* src = (const float4*)(s_prev + sbase);
        float4* dst = (float4*)sP;
        for (int i = threadIdx.x; i < (N_ * HW) / 4; i += 256) dst[i] = src[i];
    }
#endif

    // per-thread A[d, n] = -exp(log_A_real[d, n])
    float At[N_];
#pragma unroll
    for (int n = 0; n < N_; ++n) At[n] = -expf(log_A_real[d * N_ + n]);

    __syncthreads();

    const float Dp = D_param[d];
    const double ang = -(double)(v * 45) * PI_D / 180.0;
    const double cd = cos(ang), sd = sin(ang);

    for (int pp = 0; pp < 4; ++pp) {
        const int p = threadIdx.x + pp * 256;
        const int py = p >> 5, px = p & 31;

        // rotation grid in fp64 (matches numpy host-side grid construction)
        const double X = ((double)px + 0.5) / 16.0 - 1.0;
        const double Y = ((double)py + 0.5) / 16.0 - 1.0;
        const float ix = (float)((cd * X - sd * Y + 1.0) * 16.0 - 0.5);
        const float iy = (float)((sd * X + cd * Y + 1.0) * 16.0 - 0.5);

        const float x0f = floorf(ix), y0f = floorf(iy);
        const float wx1 = ix - x0f, wy1 = iy - y0f;
        const int x0 = (int)x0f, y0 = (int)y0f;
        const int x1 = x0 + 1, y1 = y0 + 1;
        const bool mx0 = (x0 >= 0) & (x0 < W_), mx1 = (x1 >= 0) & (x1 < W_);
        const bool my0 = (y0 >= 0) & (y0 < H_), my1 = (y1 >= 0) & (y1 < H_);
        const float w00 = (my0 && mx0) ? (1.f - wy1) * (1.f - wx1) : 0.f;
        const float w01 = (my0 && mx1) ? (1.f - wy1) * wx1 : 0.f;
        const float w10 = (my1 && mx0) ? wy1 * (1.f - wx1) : 0.f;
        const float w11 = (my1 && mx1) ? wy1 * wx1 : 0.f;
        const int xi0 = imin(imax(x0, 0), W_ - 1), xi1 = imin(imax(x1, 0), W_ - 1);
        const int yi0 = imin(imax(y0, 0), H_ - 1), yi1 = imin(imax(y1, 0), H_ - 1);
        const int i00 = yi0 * W_ + xi0, i01 = yi0 * W_ + xi1;
        const int i10 = yi1 * W_ + xi0, i11 = yi1 * W_ + xi1;

        const float u = u_t[((size_t)b * D_ + d) * HW + p];
        const float dt = delta[((size_t)b * D_ + d) * HW + p];
        float acc = 0.f;
        const size_t obase = sbase + p;
#pragma unroll
        for (int n = 0; n < N_; ++n) {
            const float ab = __expf(dt * At[n]);
            const float* pl = sP + n * HW;
            const float str =
                w00 * pl[i00] + w01 * pl[i01] + w10 * pl[i10] + w11 * pl[i11];
            const float bv = Bv[((size_t)b * N_ + n) * HW + p];
            const float cv = Cv[((size_t)b * N_ + n) * HW + p];
            const float sn = ab * str + (dt * bv) * u;
            out_s[obase + (size_t)n * HW] = sn;
            acc += sn * cv;
        }
        out_y[(size_t)((b * V_ + v) * D_ + d) * HW + p] = acc + u * Dp;
    }
}

// ---------------------------------------------------------------------------
extern "C" void kernel_launch(void* const* d_in, const int* in_sizes, int n_in,
                              void* d_out, int out_size, void* d_ws,
                              size_t ws_size, hipStream_t stream) {
    const float* u_t = (const float*)d_in[0];
    const float* s_prev = (const float*)d_in[1];
    const float* gn_w = (const float*)d_in[2];
    const float* gn_b = (const float*)d_in[3];
    const float* conv_delta_w = (const float*)d_in[4];
    const float* conv_delta_b = (const float*)d_in[5];
    const float* conv_B_w = (const float*)d_in[6];
    const float* conv_C_w = (const float*)d_in[7];
    const float* log_A_real = (const float*)d_in[8];
    const float* D_param = (const float*)d_in[9];
    const float* dt_bias = (const float*)d_in[10];

    float* out = (float*)d_out;
    float* out_y = out;                                      // (B,V,D,H,W)
    float* out_s = out + (size_t)B_ * V_ * D_ * HW;          // (B,V,D,N,H,W)

    float* ws = (float*)d_ws;
    float* u_norm = ws;                    // 262144
    float* delta = ws + 262144;            // 262144
    float* Bv = ws + 524288;               // 65536
    float* Cv = ws + 589824;               // 65536

    gn_kernel<<<16, 256, 0, stream>>>(u_t, gn_w, gn_b, u_norm);
    conv_wmma_kernel<<<256, 192, 0, stream>>>(u_norm, conv_delta_w,
                                              conv_delta_b, conv_B_w, conv_C_w,
                                              dt_bias, delta, Bv, Cv);
    state_kernel<<<B_ * V_ * D_, 256, N_ * HW * sizeof(float), stream>>>(
        u_t, s_prev, log_A_real, D_param, delta, Bv, Cv, out_y, out_s);
}